// ResidualVectorQuantize_19267223289862
// MI455X (gfx1250) — compile-verified
//
#include <hip/hip_runtime.h>

typedef __attribute__((ext_vector_type(2))) float v2f;
typedef __attribute__((ext_vector_type(8))) float v8f;
typedef __attribute__((ext_vector_type(4))) unsigned int v4u;
typedef __attribute__((ext_vector_type(4))) int v4i;
typedef __attribute__((ext_vector_type(8))) int v8i;

#if defined(__has_include)
#if __has_include(<hip/amd_detail/amd_gfx1250_TDM.h>)
#define RVQ_TDM_6ARG 1
#endif
#endif

#define RVQ_EPS 1e-12f
#define Bsz 8
#define Dd  1024
#define Tt  8192
#define Ncb 9
#define Kcb 1024
#define Cd  8
#define TT  64   // tokens per workgroup

// d_out flat offsets (return order: z_q, codes, latents, cl, cbl)
static constexpr size_t OFF_ZQ    = 0;
static constexpr size_t OFF_CODES = (size_t)Bsz * Dd * Tt;                 // 67108864
static constexpr size_t OFF_LAT   = OFF_CODES + (size_t)Bsz * Ncb * Tt;   // 67698688
static constexpr size_t OFF_CL    = OFF_LAT + (size_t)Bsz * Ncb * Cd * Tt;// 72417280
static constexpr size_t OFF_CBL   = OFF_CL + 1;

// LDS layout (floats)
static constexpr int L_RES = 0;          // [1024][64]
static constexpr int L_CN  = 65536;      // [1024][8] normalized codebook
static constexpr int L_CNN = 73728;      // [1024]    ||c_n||^2
static constexpr int L_ZE  = 74752;      // [8][64]
static constexpr int L_ZQ  = 75264;      // [64][8]   (straight-through z_q)
static constexpr int L_OB  = 75776;      // [1024]    out bias
static constexpr int L_IB  = 76800;      // [8]       in bias
static constexpr int L_RED = 76808;      // [256][2]
static constexpr int L_G1  = 77320;      // [128][8]  gemm1 K-split partials
static constexpr int L_TOT = 78344;      // 313376 bytes < 320KB WGP LDS

// ---- precompute weight-norm weights ----
__global__ void rvq_prep_win(const float* __restrict__ in_v,
                             const float* __restrict__ in_g,
                             float* __restrict__ ws_win) {
  int r = blockIdx.x;            // 0..143 = n*16 + c(padded)
  int n = r >> 4, c = r & 15;
  int tid = threadIdx.x;
  __shared__ float sred[256];
  float* dst = ws_win + (size_t)r * Dd;
  if (c >= Cd) {                 // zero-pad rows 8..15 for the WMMA A panel
    for (int d = tid; d < Dd; d += 256) dst[d] = 0.0f;
    return;
  }
  const float* src = in_v + (size_t)(n * Cd + c) * Dd;
  float ss = 0.0f;
  for (int d = tid; d < Dd; d += 256) { float v = src[d]; ss = fmaf(v, v, ss); }
  sred[tid] = ss; __syncthreads();
  for (int s = 128; s > 0; s >>= 1) { if (tid < s) sred[tid] += sred[tid + s]; __syncthreads(); }
  float scale = in_g[n * Cd + c] / sqrtf(sred[0]);
  for (int d = tid; d < Dd; d += 256) dst[d] = src[d] * scale;
}

__global__ void rvq_prep_wout(const float* __restrict__ out_v,
                              const float* __restrict__ out_g,
                              float* __restrict__ ws_wout) {
  int row = blockIdx.x * 256 + threadIdx.x;   // 0..9215 = n*1024 + d
  if (row >= Ncb * Dd) return;
  const float* src = out_v + (size_t)row * Cd;
  float ss = 0.0f;
#pragma unroll
  for (int c = 0; c < Cd; ++c) ss = fmaf(src[c], src[c], ss);
  float scale = out_g[row] / sqrtf(ss);
  float* dst = ws_wout + (size_t)row * Cd;
#pragma unroll
  for (int c = 0; c < Cd; ++c) dst[c] = src[c] * scale;
}

__global__ void rvq_zero_losses(float* out) {
  out[OFF_CL] = 0.0f;
  out[OFF_CBL] = 0.0f;
}

// ---- main fused RVQ kernel: one WG owns (batch b, 64-token column) ----
__global__ __launch_bounds__(256, 1)
void rvq_main(const float* __restrict__ z,
              const float* __restrict__ in_b,
              const float* __restrict__ out_b,
              const float* __restrict__ codebooks,
              const float* __restrict__ ws_win,
              const float* __restrict__ ws_wout,
              float* __restrict__ out) {
  extern __shared__ float sm[];
  float* res = sm + L_RES;
  float* cn  = sm + L_CN;
  float* cnn = sm + L_CNN;
  float* zel = sm + L_ZE;
  float* zqt = sm + L_ZQ;
  float* obd = sm + L_OB;
  float* ibd = sm + L_IB;
  float* red = sm + L_RED;
  float* g1s = sm + L_G1;

  const int tid   = threadIdx.x;
  const int lane  = tid & 31;
  const int wv    = tid >> 5;          // wave id 0..7 (uniform per wave)
  const int laneM = lane & 15;
  const int hi    = lane >> 4;         // 0: K 0,1 / rows M+0..7 ; 1: K 2,3 / rows M+8..15
  const int b     = blockIdx.y;
  const int tbase = blockIdx.x * TT;

  const int tt    = wv & 3;            // token sub-tile (16 tokens)
  const int t0    = tt * 16;
  const int half  = wv >> 2;           // K-half (gemm1) / M-half (gemm2)

  // ---- TDM: DMA the residual tile z[b, :, tbase:tbase+64] (1024x64 f32) into LDS ----
  // 2D tile: tile_dim0 = 64 contiguous elems along T, tile_dim1 = 1024 rows,
  // tensor_dim0_stride = T = 8192 elems. One descriptor, issued by wave 0 only,
  // tracked with TENSORcnt (s_wait_tensorcnt 0 before the workgroup barrier).
  if (wv == 0) {
    const unsigned long long gaddr =
        (unsigned long long)(const void*)(z + (size_t)b * Dd * Tt + tbase);
    v4u g0;
    g0.x = 1u;                                        // count = 1 (valid descriptor)
    g0.y = 0u;                                        // lds_addr = 0 (res[] base)
    g0.z = (unsigned int)(gaddr & 0xffffffffull);     // global_addr[31:0]
    g0.w = (unsigned int)((gaddr >> 32) & 0x1ffffffull) | (2u << 30); // addr[56:32], type=2
    v8i g1;
    g1[0] = 0x20000;                 // data_size = 2 (4 bytes)
    g1[1] = (int)(0x2000u << 16);    // tensor_dim0 = 8192 (bits[79:48] lo16)
    g1[2] = (int)(0x400u << 16);     // tensor_dim0 hi16 = 0 ; tensor_dim1 = 1024 lo16
    g1[3] = (int)(64u << 16);        // tensor_dim1 hi16 = 0 ; tile_dim0 = 64
    g1[4] = 0x400;                   // tile_dim1 = 1024 ; tile_dim2 = 0
    g1[5] = 8192;                    // tensor_dim0_stride[31:0] = 8192
    g1[6] = 0;                       // stride hi ; tensor_dim1_stride lo (unused, 2D)
    g1[7] = 0;
    v4i gz = {0, 0, 0, 0};
#ifdef RVQ_TDM_6ARG
    v8i gz8 = {0, 0, 0, 0, 0, 0, 0, 0};
    __builtin_amdgcn_tensor_load_to_lds(g0, g1, gz, gz, gz8, 0);
#else
    __builtin_amdgcn_tensor_load_to_lds(g0, g1, gz, gz, 0);
#endif
    __builtin_amdgcn_s_wait_tensorcnt(0);
  }

  float lsum = 0.0f;

#pragma unroll 1
  for (int n = 0; n < Ncb; ++n) {
    __syncthreads();
    // warm L2 with this stage's weight panels while we normalize the codebook
    {
      const char* pf = (const char*)(ws_win + (size_t)n * 16 * Dd);
      __builtin_prefetch(pf + tid * 256, 0, 1);
      __builtin_prefetch(pf + tid * 256 + 128, 0, 1);
      const char* pg = (const char*)(ws_wout + (size_t)n * Dd * Cd);
      __builtin_prefetch(pg + tid * 128, 0, 1);
    }
    // ---- per-stage setup: normalized codebook, biases ----
    for (int k = tid; k < Kcb; k += 256) {
      const float* cr = codebooks + (size_t)(n * Kcb + k) * Cd;
      float v[Cd]; float s2 = 0.0f;
#pragma unroll
      for (int c = 0; c < Cd; ++c) { v[c] = cr[c]; s2 = fmaf(v[c], v[c], s2); }
      float scl = 1.0f / fmaxf(sqrtf(s2), RVQ_EPS);
      float s3 = 0.0f;
#pragma unroll
      for (int c = 0; c < Cd; ++c) { float u = v[c] * scl; cn[k * Cd + c] = u; s3 = fmaf(u, u, s3); }
      cnn[k] = s3;
    }
    for (int d = tid; d < Dd; d += 256) obd[d] = out_b[n * Dd + d];
    if (tid < Cd) ibd[tid] = in_b[n * Cd + tid];
    __syncthreads();

    // ---- GEMM1: z_e[16pad x 16tok] += w_in @ residual, K=1024 split by wave pairs ----
    // Software-pipelined: iteration i+1 operand loads issue before iteration i's WMMA,
    // so the WMMA waits on loads that are a full iteration old (partial waitcnt).
    v8f acc = {};
    {
      const float* wa = ws_win + (size_t)(n * 16 + laneM) * Dd + hi * 2;
      const int dbase = half * 512;
      const int boff  = t0 + laneM;
      v2f a_c = *(const v2f*)(wa + dbase);
      v2f b_c;
      b_c.x = res[(dbase + hi * 2) * TT + boff];
      b_c.y = res[(dbase + hi * 2 + 1) * TT + boff];
#pragma unroll 4
      for (int d4 = 0; d4 < 512; d4 += 4) {
        const int dn = dbase + ((d4 + 4) & 511);     // wraps on last iter (in-bounds, branch-free)
        v2f a_n = *(const v2f*)(wa + dn);
        v2f b_n;
        b_n.x = res[(dn + hi * 2) * TT + boff];
        b_n.y = res[(dn + hi * 2 + 1) * TT + boff];
        acc = __builtin_amdgcn_wmma_f32_16x16x4_f32(false, a_c, false, b_c,
                                                    (short)0, acc, false, false);
        a_c = a_n; b_c = b_n;
      }
    }
    if (wv >= 4) {
      float* p = g1s + ((wv - 4) * 32 + lane) * 8;
#pragma unroll
      for (int r = 0; r < 8; ++r) p[r] = acc[r];
    }
    __syncthreads();
    if (wv < 4 && hi == 0) {                          // rows 0..7 are the real C channels
      const float* p = g1s + (wv * 32 + lane) * 8;
#pragma unroll
      for (int r = 0; r < 8; ++r) {
        float val = acc[r] + p[r] + ibd[r];
        zel[r * TT + t0 + laneM] = val;
        out[OFF_LAT + (size_t)(b * (Ncb * Cd) + n * Cd + r) * Tt + tbase + t0 + laneM] = val;
      }
    }
    __syncthreads();

    // ---- argmin over K=1024 (4 threads/token, wave-uniform k => LDS broadcast) ----
    {
      const int tok = tid & 63;
      const int kbase = (tid >> 6) << 8;
      float e[Cd]; float ss = 0.0f;
#pragma unroll
      for (int c = 0; c < Cd; ++c) { e[c] = zel[c * TT + tok]; ss = fmaf(e[c], e[c], ss); }
      const float scl = 1.0f / fmaxf(sqrtf(ss), RVQ_EPS);
#pragma unroll
      for (int c = 0; c < Cd; ++c) e[c] *= scl;
      float best = 3.4e38f; int bidx = kbase;
      for (int k = kbase; k < kbase + 256; ++k) {
        const float4 q0 = *(const float4*)(cn + k * Cd);
        const float4 q1 = *(const float4*)(cn + k * Cd + 4);
        float dot = 0.0f;
        dot = fmaf(e[0], q0.x, dot); dot = fmaf(e[1], q0.y, dot);
        dot = fmaf(e[2], q0.z, dot); dot = fmaf(e[3], q0.w, dot);
        dot = fmaf(e[4], q1.x, dot); dot = fmaf(e[5], q1.y, dot);
        dot = fmaf(e[6], q1.z, dot); dot = fmaf(e[7], q1.w, dot);
        const float score = fmaf(-2.0f, dot, cnn[k]);
        if (score < best) { best = score; bidx = k; }   // strict < : first-min tie break
      }
      red[tid * 2] = best; red[tid * 2 + 1] = (float)bidx;
    }
    __syncthreads();
    if (tid < 64) {                                   // combine 4 K-ranges, lowest k wins ties
      const int tok = tid;
      float bb = red[tid * 2]; int bi = (int)red[tid * 2 + 1];
      for (int j = 1; j < 4; ++j) {
        const float s = red[(j * 64 + tid) * 2];
        if (s < bb) { bb = s; bi = (int)red[(j * 64 + tid) * 2 + 1]; }
      }
      out[OFF_CODES + (size_t)(b * Ncb + n) * Tt + tbase + tok] = (float)bi;
      const float* cr = codebooks + (size_t)(n * Kcb + bi) * Cd;
#pragma unroll
      for (int c = 0; c < Cd; ++c) {
        const float ze = zel[c * TT + tok];
        const float zq = cr[c];
        const float df = ze - zq;
        lsum = fmaf(df, df, lsum);                    // commit == cb loss numerically
        zqt[tok * Cd + c] = ze + (zq - ze);           // straight-through, as in reference
      }
    }
    __syncthreads();

    // ---- GEMM2: residual -= w_out @ z_q + out_b ; 8 waves x 32 M-tiles ----
    // A-panel loads software-pipelined one tile ahead.
    {
      const v2f b0 = *(const v2f*)(zqt + (t0 + laneM) * Cd + 0 + hi * 2); // B K=0..3
      const v2f b4 = *(const v2f*)(zqt + (t0 + laneM) * Cd + 4 + hi * 2); // B K=4..7
      const int mbase = half * 512;
      const float* wb = ws_wout + (size_t)(n * Dd + mbase + laneM) * Cd + hi * 2;
      v2f a0_c = *(const v2f*)(wb);
      v2f a4_c = *(const v2f*)(wb + 4);
#pragma unroll 2
      for (int i = 0; i < 32; ++i) {
        const int inext = (i + 1) & 31;               // wraps on last iter (in-bounds)
        const float* wn = wb + (size_t)inext * 16 * Cd;
        v2f a0_n = *(const v2f*)(wn);
        v2f a4_n = *(const v2f*)(wn + 4);
        v8f c2 = {};
        c2 = __builtin_amdgcn_wmma_f32_16x16x4_f32(false, a0_c, false, b0, (short)0, c2, false, false);
        c2 = __builtin_amdgcn_wmma_f32_16x16x4_f32(false, a4_c, false, b4, (short)0, c2, false, false);
        const int rbase = mbase + i * 16 + hi * 8;
#pragma unroll
        for (int r = 0; r < 8; ++r) {
          const int row = rbase + r;
          res[row * TT + t0 + laneM] -= c2[r] + obd[row];
        }
        a0_c = a0_n; a4_c = a4_n;
      }
    }
  }
  __syncthreads();

  // ---- outputs: z_q_out = z - residual_final ----
  {
    const int tcol = tid & 63, drow = tid >> 6;
    for (int dd = drow; dd < Dd; dd += 4) {
      const size_t g = (size_t)(b * Dd + dd) * Tt + tbase + tcol;
      out[OFF_ZQ + g] = z[g] - res[dd * TT + tcol];
    }
  }
  // ---- loss reduction ----
  red[tid] = lsum;
  __syncthreads();
  if (tid == 0) {
    float s = 0.0f;
    for (int i = 0; i < 256; ++i) s += red[i];
    s *= (1.0f / (float)((size_t)Bsz * Cd * Tt));     // mean over [B,C,T], summed over stages
    atomicAdd(&out[OFF_CL], s);
    atomicAdd(&out[OFF_CBL], s);
  }
}

extern "C" void kernel_launch(void* const* d_in, const int* in_sizes, int n_in,
                              void* d_out, int out_size, void* d_ws, size_t ws_size,
                              hipStream_t stream) {
  const float* z     = (const float*)d_in[0];
  const float* in_v  = (const float*)d_in[1];
  const float* in_g  = (const float*)d_in[2];
  const float* in_b  = (const float*)d_in[3];
  const float* out_v = (const float*)d_in[4];
  const float* out_g = (const float*)d_in[5];
  const float* out_b = (const float*)d_in[6];
  const float* cb    = (const float*)d_in[7];
  float* out = (float*)d_out;

  float* ws_win  = (float*)d_ws;                   // [9][16][1024] zero-padded A panel
  float* ws_wout = ws_win + (size_t)Ncb * 16 * Dd; // [9][1024][8]

  const size_t smem = (size_t)L_TOT * sizeof(float);   // ~306 KB (CDNA5: 320 KB/WG)
  (void)hipFuncSetAttribute(reinterpret_cast<const void*>(rvq_main),
                            hipFuncAttributeMaxDynamicSharedMemorySize, (int)smem);

  rvq_prep_win <<<Ncb * 16, 256, 0, stream>>>(in_v, in_g, ws_win);
  rvq_prep_wout<<<(Ncb * Dd + 255) / 256, 256, 0, stream>>>(out_v, out_g, ws_wout);
  rvq_zero_losses<<<1, 1, 0, stream>>>(out);
  rvq_main<<<dim3(Tt / TT, Bsz), 256, smem, stream>>>(z, in_b, out_b, cb,
                                                      ws_win, ws_wout, out);
}